// GNN_Conv_85959475462176
// MI455X (gfx1250) — compile-verified
//
#include <hip/hip_runtime.h>

#define HIDDEN 128
#define LEAKY 0.01f
#define BN_EPS 1e-5f

typedef float v2f __attribute__((ext_vector_type(2)));
typedef float v8f __attribute__((ext_vector_type(8)));

// ---------------------------------------------------------------- utilities
__global__ void zero_kernel(float* p, int n) {
  int i = blockIdx.x * blockDim.x + threadIdx.x;
  int stride = gridDim.x * blockDim.x;
  for (; i < n; i += stride) p[i] = 0.0f;
}

__global__ void count_kernel(const int* __restrict__ dst, float* cnt, int E) {
  int e = blockIdx.x * blockDim.x + threadIdx.x;
  if (e < E) unsafeAtomicAdd(&cnt[dst[e]], 1.0f);
}

__global__ void invcnt_kernel(float* cnt, int n) {
  int i = blockIdx.x * blockDim.x + threadIdx.x;
  if (i < n) cnt[i] = 1.0f / fmaxf(cnt[i], 1.0f);
}

// ------------------------------------------------------- edge scatter (mean agg numerator)
// 32 threads per edge: coalesced float4 gather of the src row, 4 f32 atomics to dst row.
__global__ void scatter_kernel(const float* __restrict__ h,
                               const int* __restrict__ src,
                               const int* __restrict__ dst,
                               float* agg, int E) {
  long long tid = (long long)blockIdx.x * blockDim.x + threadIdx.x;
  int e = (int)(tid >> 5);
  if (e >= E) return;
  int f = ((int)tid & 31) * 4;
  int s = src[e];
  int d = dst[e];
  const float4 v = *(const float4*)(h + (size_t)s * HIDDEN + f);
  float* p = agg + (size_t)d * HIDDEN + f;
  unsafeAtomicAdd(p + 0, v.x);
  unsafeAtomicAdd(p + 1, v.y);
  unsafeAtomicAdd(p + 2, v.z);
  unsafeAtomicAdd(p + 3, v.w);
}

// ------------------------------------------------------- fused SAGE layer GEMM
// Weights are staged into LDS in WMMA-fragment order so the inner loop issues a
// single conflict-free ds_load_b64 per B fragment (no register repacking):
//   ldsB float2 index (K4*128 + col)*2 + half  =  { W[4*K4+2*half][col], W[4*K4+2*half+1][col] }
__device__ __forceinline__ void stage_weights(const float* __restrict__ W,
                                              float* ldsf, int tid) {
#pragma unroll 4
  for (int K4 = 0; K4 < 32; ++K4) {
    const float* wrow = W + K4 * 4 * HIDDEN + tid;   // tid = column 0..127
    float r0 = wrow[0 * HIDDEN];
    float r1 = wrow[1 * HIDDEN];
    float r2 = wrow[2 * HIDDEN];
    float r3 = wrow[3 * HIDDEN];
    float* d = ldsf + (size_t)(K4 * HIDDEN + tid) * 4;
    v2f p0; p0.x = r0; p0.y = r1;                    // half = 0 fragment
    v2f p1; p1.x = r2; p1.y = r3;                    // half = 1 fragment
    *(v2f*)(d + 0) = p0;
    *(v2f*)(d + 2) = p1;
  }
}

template <bool SCALE>
__device__ __forceinline__ void gemm_phase(const float* ldsf,
                                           const float* __restrict__ arow,
                                           float scale, int lanelo, int half,
                                           v8f c[8]) {
#pragma unroll 2
  for (int K4 = 0; K4 < 32; ++K4) {
    float2 av = *(const float2*)(arow + K4 * 4);     // A 16x4: v0=K{0|2}, v1=K{1|3}
    v2f a;
    if (SCALE) { a.x = av.x * scale; a.y = av.y * scale; }
    else       { a.x = av.x;         a.y = av.y; }
#pragma unroll
    for (int t = 0; t < 8; ++t) {
      const int fi = ((K4 * HIDDEN + t * 16 + lanelo) * 2 + half) * 2;
      v2f b = *(const v2f*)(ldsf + fi);              // one ds_load_b64, bank-conflict-free
      c[t] = __builtin_amdgcn_wmma_f32_16x16x4_f32(false, a, false, b,
                                                   (short)0, c[t], false, false);
    }
  }
}

// out = (agg * invc) @ Wl + h @ Wr + bl, LeakyReLU.  Block = 4 waves, tile 64x128.
__global__ __launch_bounds__(128) void sage_gemm_kernel(
    const float* __restrict__ agg, const float* __restrict__ invc,
    const float* __restrict__ hin,
    const float* __restrict__ Wl, const float* __restrict__ Wr,
    const float* __restrict__ bl, float* __restrict__ out, int M) {
  __shared__ float ldsf[HIDDEN * HIDDEN];            // 64 KB fragment-order weight stage

  const int tid = threadIdx.x;
  const int lane = tid & 31;
  const int wave = tid >> 5;
  const int lanelo = lane & 15;
  const int half = lane >> 4;                        // 0: lanes 0-15, 1: lanes 16-31
  const int rowbase = blockIdx.x * 64 + wave * 16;
  int r = rowbase + lanelo;
  if (r >= M) r = M - 1;                             // uniform flow; guard only stores
  const float ic = invc[r];

  v8f c[8] = {};                                     // 8 col-tiles x 8 accum VGPRs (f32)

  // ---- phase 1: (agg/deg) @ Wl ----
  stage_weights(Wl, ldsf, tid);
  __syncthreads();
  gemm_phase<true>(ldsf, agg + (size_t)r * HIDDEN + half * 2, ic, lanelo, half, c);
  __syncthreads();

  // ---- phase 2: h @ Wr ----
  stage_weights(Wr, ldsf, tid);
  __syncthreads();
  gemm_phase<false>(ldsf, hin + (size_t)r * HIDDEN + half * 2, 1.0f, lanelo, half, c);

  // ---- epilogue: +bias, LeakyReLU, store (C/D layout: VGPR v -> rows v, v+8) ----
#pragma unroll
  for (int t = 0; t < 8; ++t) {
    const float bias = bl[t * 16 + lanelo];
#pragma unroll
    for (int v = 0; v < 8; ++v) {
      const int row = rowbase + v + half * 8;
      if (row < M) {
        float y = c[t][v] + bias;
        y = (y >= 0.0f) ? y : LEAKY * y;
        out[(size_t)row * HIDDEN + t * 16 + lanelo] = y;
      }
    }
  }
}

// ------------------------------------------------------- batch-norm (train mode)
__global__ __launch_bounds__(128) void bnstats_kernel(const float* __restrict__ y,
                                                      float* stats, int M) {
  const int t = threadIdx.x;                         // feature column 0..127
  int r0 = blockIdx.x * 128;
  int rend = r0 + 128;
  if (rend > M) rend = M;
  float s = 0.0f, s2 = 0.0f;
  for (int r = r0; r < rend; ++r) {
    float v = y[(size_t)r * HIDDEN + t];
    s += v;
    s2 += v * v;
  }
  unsafeAtomicAdd(&stats[t], s);
  unsafeAtomicAdd(&stats[HIDDEN + t], s2);
}

__global__ void bnfinal_kernel(const float* __restrict__ stats,
                               const float* __restrict__ gamma,
                               const float* __restrict__ beta,
                               float* ss, int M) {
  const int t = threadIdx.x;
  const float invM = 1.0f / (float)M;
  float mu = stats[t] * invM;
  float var = stats[HIDDEN + t] * invM - mu * mu;    // biased variance
  var = fmaxf(var, 0.0f);
  float sc = gamma[t] * rsqrtf(var + BN_EPS);
  ss[t] = sc;
  ss[HIDDEN + t] = beta[t] - mu * sc;
}

__global__ void bnapply_kernel(float* y, const float* __restrict__ ss, int n4) {
  int i = blockIdx.x * blockDim.x + threadIdx.x;
  if (i >= n4) return;
  float4 v = ((float4*)y)[i];
  int c = (i * 4) & (HIDDEN - 1);
  v.x = v.x * ss[c + 0] + ss[HIDDEN + c + 0];
  v.y = v.y * ss[c + 1] + ss[HIDDEN + c + 1];
  v.z = v.z * ss[c + 2] + ss[HIDDEN + c + 2];
  v.w = v.w * ss[c + 3] + ss[HIDDEN + c + 3];
  ((float4*)y)[i] = v;
}

// ---------------------------------------------------------------- launcher
extern "C" void kernel_launch(void* const* d_in, const int* in_sizes, int n_in,
                              void* d_out, int out_size, void* d_ws, size_t ws_size,
                              hipStream_t stream) {
  const float* x     = (const float*)d_in[0];
  const int*   edge  = (const int*)d_in[1];
  const float* Wl    = (const float*)d_in[2];
  const float* bl    = (const float*)d_in[3];
  const float* Wr    = (const float*)d_in[4];
  const float* gamma = (const float*)d_in[5];
  const float* beta  = (const float*)d_in[6];

  const int N = in_sizes[0] / HIDDEN;   // 50000 nodes
  const int E = in_sizes[1] / 2;        // 800000 edges
  const int* src = edge;
  const int* dst = edge + E;

  float* ws = (float*)d_ws;
  const size_t NH = (size_t)N * HIDDEN;
  float* hA    = ws;
  float* hB    = hA + NH;
  float* agg   = hB + NH;
  float* cnt   = agg + NH;
  float* stats = cnt + (size_t)((N + 255) & ~255);
  float* ss    = stats + 2 * HIDDEN;

  // degree counts + reciprocals (layer-invariant)
  zero_kernel<<<256, 256, 0, stream>>>(cnt, N);
  count_kernel<<<(E + 255) / 256, 256, 0, stream>>>(dst, cnt, E);
  invcnt_kernel<<<(N + 255) / 256, 256, 0, stream>>>(cnt, N);

  const float* hin = x;
  float* bufs[4] = {hA, hB, hA, (float*)d_out};
  for (int l = 0; l < 4; ++l) {
    float* hout = bufs[l];

    zero_kernel<<<2048, 256, 0, stream>>>(agg, (int)NH);

    long long sthreads = (long long)E * 32;
    int sblocks = (int)((sthreads + 255) / 256);
    scatter_kernel<<<sblocks, 256, 0, stream>>>(hin, src, dst, agg, E);

    sage_gemm_kernel<<<(N + 63) / 64, 128, 0, stream>>>(
        agg, cnt, hin,
        Wl + (size_t)l * HIDDEN * HIDDEN, Wr + (size_t)l * HIDDEN * HIDDEN,
        bl + l * HIDDEN, hout, N);

    zero_kernel<<<1, 256, 0, stream>>>(stats, 2 * HIDDEN);
    bnstats_kernel<<<(N + 127) / 128, 128, 0, stream>>>(hout, stats, N);
    bnfinal_kernel<<<1, 128, 0, stream>>>(stats, gamma + l * HIDDEN,
                                          beta + l * HIDDEN, ss, N);
    bnapply_kernel<<<((int)(NH / 4) + 255) / 256, 256, 0, stream>>>(
        hout, ss, (int)(NH / 4));

    hin = hout;
  }
}